// ConceptHierarchyModule_47665547051323
// MI455X (gfx1250) — compile-verified
//
#include <hip/hip_runtime.h>

typedef float v2f __attribute__((ext_vector_type(2)));
typedef float v8f __attribute__((ext_vector_type(8)));

#define FDIM 128

// ---------------- init: counts = 0, idx[] = -1 ----------------
__global__ void chm_init_kernel(int* __restrict__ counts, int* __restrict__ idx, long long tot) {
    long long t = (long long)blockIdx.x * blockDim.x + threadIdx.x;
    if (t < 16) counts[t] = 0;
    if (t < tot) idx[t] = -1;
}

// ---------------- bucket nodes by level (counting sort) ----------------
__global__ void chm_bucket_kernel(const int* __restrict__ levels, int* __restrict__ counts,
                                  int* __restrict__ idx, int N, int L, int Ncap) {
    int n = blockIdx.x * blockDim.x + threadIdx.x;
    if (n >= N) return;
    int l = levels[n];
    if (l >= 0 && l < L) {
        int pos = atomicAdd(&counts[l], 1);
        idx[l * Ncap + pos] = n;
    }
}

// ---------------- per-node transform: Y[n] = 0.2*(X[n] @ W[l]^T + b[l]) ----------------
// One wave = 16 nodes (one level bucket tile) x 128 output features.
// Uses V_WMMA_F32_16X16X4_F32, K accumulated 128/4 = 32 steps, 8 output tiles.
__global__ __launch_bounds__(128) void chm_gemm_kernel(
    const float* __restrict__ X, const float* __restrict__ W, const float* __restrict__ Bv,
    const int* __restrict__ counts, const int* __restrict__ idx,
    float* __restrict__ Y, int Ncap, int L) {

    const int lane  = threadIdx.x & 31;
    const int wave  = threadIdx.x >> 5;
    const int group = blockIdx.x * 4 + wave;
    const int groupsPerLevel = Ncap >> 4;

    const int level = group / groupsPerLevel;
    const int gi    = group - level * groupsPerLevel;
    if (level >= L) return;                 // wave-uniform exit
    const int cnt = counts[level];
    if (gi * 16 >= cnt) return;             // wave-uniform exit

    const int m     = lane & 15;            // row within tile (for A) / col within tile (for B)
    const int khalf = lane >> 4;            // 0: K pair {0,1}; 1: K pair {2,3}
    const int base  = level * Ncap + gi * 16;

    int na = idx[base + m];
    if (na < 0) na = 0;                     // padded slot: load row 0, store suppressed later

    const float* __restrict__ Wl   = W + (size_t)level * FDIM * FDIM;
    const float* __restrict__ xrow = X + (size_t)na * FDIM + 2 * khalf;
    const float* __restrict__ wcol = Wl + 2 * khalf;

    v8f zero = {};
    v8f acc[8];
#pragma unroll
    for (int j = 0; j < 8; ++j) acc[j] = zero;

    for (int k0 = 0; k0 < FDIM; k0 += 4) {
        v2f a = *(const v2f*)(xrow + k0);   // A: X[m][k0+2*khalf .. +1]
#pragma unroll
        for (int j = 0; j < 8; ++j) {
            // B: W[n0 + m][k0 + 2*khalf .. +1]  (B = W^T tile)
            v2f b = *(const v2f*)(wcol + (size_t)(j * 16 + m) * FDIM + k0);
            acc[j] = __builtin_amdgcn_wmma_f32_16x16x4_f32(
                false, a, false, b, (short)0, acc[j], false, false);
        }
    }

    // node indices for the 8 D-rows this lane owns (rows khalf*8 + v)
    int nn[8];
#pragma unroll
    for (int v = 0; v < 8; ++v) nn[v] = idx[base + 8 * khalf + v];

    const float* __restrict__ bl = Bv + level * FDIM;
#pragma unroll
    for (int j = 0; j < 8; ++j) {
        float bias = bl[j * 16 + m];
#pragma unroll
        for (int v = 0; v < 8; ++v) {
            if (nn[v] >= 0)
                Y[(size_t)nn[v] * FDIM + j * 16 + m] = 0.2f * (acc[j][v] + bias);
        }
    }
}

// ---------------- out = node_features ----------------
__global__ void chm_copy_kernel(const float* __restrict__ X, float* __restrict__ out, long long n4) {
    long long t = (long long)blockIdx.x * blockDim.x + threadIdx.x;
    if (t < n4) ((float4*)out)[t] = ((const float4*)X)[t];
}

// ---------------- edge scatter: out[hi] += Y[lo] for valid edges ----------------
// One wave per edge; each lane handles 4 consecutive floats (32*4 = 128).
__global__ __launch_bounds__(256) void chm_edge_kernel(
    const int* __restrict__ edges, const int* __restrict__ levels,
    const float* __restrict__ Y, float* __restrict__ out, int E, int L) {
    long long t = (long long)blockIdx.x * blockDim.x + threadIdx.x;
    int e    = (int)(t >> 5);
    int lane = (int)(t & 31);
    if (e >= E) return;
    int lo = edges[2 * e + 0];
    int hi = edges[2 * e + 1];
    int ll = levels[lo];
    int hl = levels[hi];
    if (ll < 0 || ll >= L || hl <= ll) return;   // wave-uniform (same e per wave)
    float4 v = *(const float4*)(Y + (size_t)lo * FDIM + lane * 4);
    float* o = out + (size_t)hi * FDIM + lane * 4;
    atomicAdd(o + 0, v.x);
    atomicAdd(o + 1, v.y);
    atomicAdd(o + 2, v.z);
    atomicAdd(o + 3, v.w);
}

extern "C" void kernel_launch(void* const* d_in, const int* in_sizes, int n_in,
                              void* d_out, int out_size, void* d_ws, size_t ws_size,
                              hipStream_t stream) {
    const float* X      = (const float*)d_in[0];   // (N, 128)
    const float* W      = (const float*)d_in[1];   // (L, 128, 128)
    const float* Bv     = (const float*)d_in[2];   // (L, 128)
    const int*   levels = (const int*)d_in[3];     // (N,)
    const int*   edges  = (const int*)d_in[4];     // (E, 2)
    float*       out    = (float*)d_out;           // (N, 128)

    const int N    = in_sizes[0] / FDIM;
    const int L    = in_sizes[1] / (FDIM * FDIM);
    const int E    = in_sizes[4] / 2;
    const int Ncap = ((N + 15) / 16) * 16;

    char* ws     = (char*)d_ws;
    int*  counts = (int*)ws;                                    // 16 ints
    int*  idx    = (int*)(ws + 64);                             // L*Ncap ints
    size_t yoff  = (64 + (size_t)L * Ncap * 4 + 255) & ~(size_t)255;
    float* Y     = (float*)(ws + yoff);                         // N*128 floats

    // 1) init buckets
    {
        long long tot = (long long)L * Ncap;
        int blocks = (int)((tot + 255) / 256);
        chm_init_kernel<<<blocks, 256, 0, stream>>>(counts, idx, tot);
    }
    // 2) bucket nodes by level
    chm_bucket_kernel<<<(N + 255) / 256, 256, 0, stream>>>(levels, counts, idx, N, L, Ncap);
    // 3) WMMA transform -> Y
    {
        int groups = L * (Ncap >> 4);
        int blocks = (groups + 3) / 4;   // 4 waves / block
        chm_gemm_kernel<<<blocks, 128, 0, stream>>>(X, W, Bv, counts, idx, Y, Ncap, L);
    }
    // 4) out = node_features
    {
        long long n4 = (long long)N * FDIM / 4;
        int blocks = (int)((n4 + 255) / 256);
        chm_copy_kernel<<<blocks, 256, 0, stream>>>(X, out, n4);
    }
    // 5) edge scatter with f32 atomics
    {
        long long threadsTot = (long long)E * 32;
        int blocks = (int)((threadsTot + 255) / 256);
        chm_edge_kernel<<<blocks, 256, 0, stream>>>(edges, levels, Y, out, E, L);
    }
}